// SpatialTransformer_42099269435502
// MI455X (gfx1250) — compile-verified
//
#include <hip/hip_runtime.h>
#include <stdint.h>

// ---- problem geometry (fixed by the reference) ----
#define DD 160
#define HH 192
#define WW 160
#define HW (HH * WW)            // 30720
#define DHW (DD * HH * WW)      // 4915200
#define NBATCH 2
#define QPB (DHW / 4)           // quads (4 voxels) per batch = 1228800
#define CHUNK 256               // quads per chunk = one block-wide tile (1024 voxels)
#define TOTAL_CHUNKS ((NBATCH * QPB) / CHUNK)  // 9600
#define NBLOCKS 1200            // persistent blocks; 8 chunks each -> async pipeline depth

// Native 4-float vector (usable with __builtin_nontemporal_store)
typedef float v4f_t __attribute__((vector_size(4 * sizeof(float))));

// ---- CDNA5 async-to-LDS path (guarded; fallback = direct b128 loads) ----
#if defined(__gfx1250__) &&                                            \
    __has_builtin(__builtin_amdgcn_global_load_async_to_lds_b128) &&   \
    __has_builtin(__builtin_amdgcn_s_wait_asynccnt)
#define USE_ASYNC 1
#else
#define USE_ASYNC 0
#endif

static __device__ __forceinline__ int clampi(int v, int lo, int hi) {
    return v < lo ? lo : (v > hi ? hi : v);
}

// Decode global chunk id + lane -> (batch, linear voxel index of this lane's quad)
static __device__ __forceinline__ void chunk_decode(int c, int tid, int& b, int& lin) {
    int gq = c * CHUNK + tid;
    b = (gq >= QPB) ? 1 : 0;
    lin = (gq - b * QPB) * 4;
}

// Trilinear sample, zero padding. Out-of-range validity is folded into the
// per-axis weights, then the 8-corner weighted sum is factored into
// x-lerp -> y-lerp -> z-lerp (exact same algebra, 14 mul/fma instead of 24).
static __device__ __forceinline__ float sample1(const float* __restrict__ s,
                                                float pz, float py, float px) {
    float z0f = floorf(pz), y0f = floorf(py), x0f = floorf(px);
    float wz1 = pz - z0f, wy1 = py - y0f, wx1 = px - x0f;
    int z0 = (int)z0f, y0 = (int)y0f, x0 = (int)x0f;
    int z1 = z0 + 1, y1 = y0 + 1, x1 = x0 + 1;
    float az0 = (z0 >= 0 && z0 < DD) ? (1.0f - wz1) : 0.0f;
    float az1 = (z1 >= 0 && z1 < DD) ? wz1 : 0.0f;
    float ay0 = (y0 >= 0 && y0 < HH) ? (1.0f - wy1) : 0.0f;
    float ay1 = (y1 >= 0 && y1 < HH) ? wy1 : 0.0f;
    float ax0 = (x0 >= 0 && x0 < WW) ? (1.0f - wx1) : 0.0f;
    float ax1 = (x1 >= 0 && x1 < WW) ? wx1 : 0.0f;
    int zc0 = clampi(z0, 0, DD - 1), zc1 = clampi(z1, 0, DD - 1);
    int yc0 = clampi(y0, 0, HH - 1), yc1 = clampi(y1, 0, HH - 1);
    int xc0 = clampi(x0, 0, WW - 1), xc1 = clampi(x1, 0, WW - 1);
    int r00 = (zc0 * HH + yc0) * WW, r01 = (zc0 * HH + yc1) * WW;
    int r10 = (zc1 * HH + yc0) * WW, r11 = (zc1 * HH + yc1) * WW;
    float v000 = s[r00 + xc0], v001 = s[r00 + xc1];
    float v010 = s[r01 + xc0], v011 = s[r01 + xc1];
    float v100 = s[r10 + xc0], v101 = s[r10 + xc1];
    float v110 = s[r11 + xc0], v111 = s[r11 + xc1];
    float x00 = fmaf(ax1, v001, ax0 * v000);
    float x01 = fmaf(ax1, v011, ax0 * v010);
    float x10 = fmaf(ax1, v101, ax0 * v100);
    float x11 = fmaf(ax1, v111, ax0 * v110);
    float yA = fmaf(ay1, x01, ay0 * x00);
    float yB = fmaf(ay1, x11, ay0 * x10);
    return fmaf(az1, yB, az0 * yA);
}

// Compute 4 consecutive voxels of one chunk and store a b128 (non-temporal:
// output is write-once, keep L2 for the src gather working set).
static __device__ __forceinline__ void compute_chunk(const float* __restrict__ src,
                                                     float* __restrict__ out,
                                                     int c, int tid,
                                                     float4 f0, float4 f1, float4 f2) {
    int b, lin;
    chunk_decode(c, tid, b, lin);
    int d = lin / HW;
    int rem = lin - d * HW;
    int h = rem / WW;
    int w = rem - h * WW;
    const float* sb = src + (size_t)b * DHW;
    float fd = (float)d, fh = (float)h, fw = (float)w;
    v4f_t o;
    o[0] = sample1(sb, fd + f0.x, fh + f1.x, fw + 0.0f + f2.x);
    o[1] = sample1(sb, fd + f0.y, fh + f1.y, fw + 1.0f + f2.y);
    o[2] = sample1(sb, fd + f0.z, fh + f1.z, fw + 2.0f + f2.z);
    o[3] = sample1(sb, fd + f0.w, fh + f1.w, fw + 3.0f + f2.w);
    __builtin_nontemporal_store(o, (v4f_t*)(out + (size_t)b * DHW + lin));
}

// Prefetch the next chunk's src identity neighborhood (gathers cluster near it).
static __device__ __forceinline__ void prefetch_src(const float* src, int c, int tid) {
    int b, lin;
    chunk_decode(c, tid, b, lin);
    const float* sp = src + (size_t)b * DHW + lin;
    __builtin_prefetch(sp, 0, 3);                       // own z-plane line
    if (lin + HW < DHW) __builtin_prefetch(sp + HW, 0, 3);  // z+1 plane line
}

#if USE_ASYNC
// Builtin prototype (clang-22): (int4 AS1*, int4 AS3*, imm offset, imm cpol)
// with GCC-style vector_size(16) int pointee.
typedef int v4i_t __attribute__((vector_size(4 * sizeof(int))));
typedef __attribute__((address_space(1))) v4i_t gv4i_t;
typedef __attribute__((address_space(3))) v4i_t lv4i_t;

static __device__ __forceinline__ gv4i_t* to_as1(const void* p) {
    return (gv4i_t*)(unsigned long long)p;              // flat global addr == AS1 addr
}
static __device__ __forceinline__ lv4i_t* to_as3(void* p) {
    // HW truncates flat LDS addresses to addr[31:0]; do the same.
    return (lv4i_t*)(unsigned int)(unsigned long long)p;
}

// Issue 3x async b128 copies (flow z/y/x channels) for chunk c into LDS buffer fb.
static __device__ __forceinline__ void stage_chunk(const float* flow, const float* src,
                                                   float4 (*fb)[CHUNK], int c, int tid) {
    int b, lin;
    chunk_decode(c, tid, b, lin);
    const float* fp = flow + (size_t)(3 * b) * DHW + lin;
    __builtin_amdgcn_global_load_async_to_lds_b128(to_as1(fp),           to_as3(&fb[0][tid]), 0, 0);
    __builtin_amdgcn_global_load_async_to_lds_b128(to_as1(fp + DHW),     to_as3(&fb[1][tid]), 0, 0);
    __builtin_amdgcn_global_load_async_to_lds_b128(to_as1(fp + 2 * DHW), to_as3(&fb[2][tid]), 0, 0);
    prefetch_src(src, c, tid);
}
#endif

__global__ __launch_bounds__(256) void st3d_warp_kernel(const float* __restrict__ src,
                                                        const float* __restrict__ flow,
                                                        float* __restrict__ out) {
    const int tid = threadIdx.x;

#if USE_ASYNC
    // Double-buffered flow staging: 2 x 3 channels x 256 lanes x 16B = 24 KB LDS.
    __shared__ float4 fbuf[2][3][CHUNK];
    int c = (int)blockIdx.x;
    int buf = 0;
    if (c < TOTAL_CHUNKS) stage_chunk(flow, src, fbuf[0], c, tid);
    for (; c < TOTAL_CHUNKS; c += (int)gridDim.x) {
        int cn = c + (int)gridDim.x;
        if (cn < TOTAL_CHUNKS) {
            stage_chunk(flow, src, fbuf[buf ^ 1], cn, tid);   // overlap next chunk's copy
            __builtin_amdgcn_s_wait_asynccnt(3);              // in-order: current chunk landed
        } else {
            __builtin_amdgcn_s_wait_asynccnt(0);
        }
        asm volatile("" ::: "memory");  // keep LDS reads below the asynccnt wait
        float4 f0 = fbuf[buf][0][tid];  // each lane consumes its own slot -> no barrier
        float4 f1 = fbuf[buf][1][tid];
        float4 f2 = fbuf[buf][2][tid];
        compute_chunk(src, out, c, tid, f0, f1, f2);
        buf ^= 1;
    }
#else
    for (int c = (int)blockIdx.x; c < TOTAL_CHUNKS; c += (int)gridDim.x) {
        int b, lin;
        chunk_decode(c, tid, b, lin);
        const float* fp = flow + (size_t)(3 * b) * DHW + lin;
        float4 f0 = *(const float4*)(fp);
        float4 f1 = *(const float4*)(fp + DHW);
        float4 f2 = *(const float4*)(fp + 2 * DHW);
        int cn = c + (int)gridDim.x;
        if (cn < TOTAL_CHUNKS) prefetch_src(src, cn, tid);
        compute_chunk(src, out, c, tid, f0, f1, f2);
    }
#endif
}

extern "C" void kernel_launch(void* const* d_in, const int* in_sizes, int n_in,
                              void* d_out, int out_size, void* d_ws, size_t ws_size,
                              hipStream_t stream) {
    const float* src  = (const float*)d_in[0];   // [2,1,160,192,160] f32
    const float* flow = (const float*)d_in[1];   // [2,3,160,192,160] f32
    float* out = (float*)d_out;                  // [2,1,160,192,160] f32
    (void)in_sizes; (void)n_in; (void)out_size; (void)d_ws; (void)ws_size;
    st3d_warp_kernel<<<dim3(NBLOCKS), dim3(256), 0, stream>>>(src, flow, out);
}